// MDNLossNegative_34789235097871
// MI455X (gfx1250) — compile-verified
//
#include <hip/hip_runtime.h>
#include <hip/hip_bf16.h>
#include <math.h>

typedef __attribute__((ext_vector_type(2))) float v2f;
typedef __attribute__((ext_vector_type(8))) float v8f;

#define BS_   64
#define W_    8
#define K_    10
#define N_    4096
#define P_    16
#define T_    20
#define LOG2PI_ 1.837877066409345483560659472811f

// One workgroup per (b, w). 256 threads = 8 wave32.
__global__ __launch_bounds__(256) void mdn_loss_main(
    const float* __restrict__ mu,
    const float* __restrict__ sigma,
    const float* __restrict__ pi,
    const float* __restrict__ neg,
    const float* __restrict__ pos,
    const int*  __restrict__ flip_p,
    const int*  __restrict__ cap_p,
    float* __restrict__ partial)
{
  __shared__ float    s_dist[N_];
  __shared__ float    s_pos2[16];
  __shared__ float    s_red[256];
  __shared__ int      s_scan[256];
  __shared__ float    s_mu0[K_], s_mu1[K_], s_is0[K_], s_is1[K_], s_ck[K_];
  __shared__ float    s_dmin, s_lse, s_S;
  __shared__ unsigned s_cnt;

  const int bw   = blockIdx.x;
  const int b    = bw / W_;
  const int w    = bw % W_;
  const int tid  = threadIdx.x;
  const int lane = tid & 31;
  const int wave = tid >> 5;

  const int wp[8] = {0, 2, 4, 6, 9, 12, 15, 19};
  const int twp = wp[w];

  const float2* posv = (const float2*)pos;
  const float2* negv = (const float2*)neg;

  // ---- A fragment: 16 positive waypoints (M=16 rows, K=2 valid of 4) ----
  // Layout (f32 16x4 A): lanes 0-15 hold K=0,1 for M=lane; lanes 16-31 hold
  // K=2,3 (zero-padded).  The wh-flip cancels in the distance, so no flip here.
  float ax = 0.0f, ay = 0.0f;
  if (lane < 16) {
    float2 q = posv[(b * P_ + lane) * T_ + twp];
    ax = q.x; ay = q.y;
    if (wave == 0) s_pos2[lane] = ax * ax + ay * ay;
  }
  __syncthreads();

  const int half = (lane >= 16) ? 8 : 0;
  float p2[8];
#pragma unroll
  for (int r = 0; r < 8; ++r) p2[r] = s_pos2[half + r];

  v2f A = {ax, ay};

  // ---- phase 1: min-over-P distances for all N negatives via WMMA ----
  float local_min = 3.0e38f;
  for (int tile = wave; tile < N_ / 16; tile += 8) {
    const int n0 = tile * 16;
    float bx = 0.0f, by = 0.0f, n2 = 0.0f;
    if (lane < 16) {
      float2 q = negv[b * N_ + n0 + lane];
      bx = q.x; by = q.y; n2 = bx * bx + by * by;
    }
    v2f Bf = {bx, by};
    v8f C = {};
    // D[M,N] = pos_M . neg_N   (K padded 2->4 with zeros)
    C = __builtin_amdgcn_wmma_f32_16x16x4_f32(false, A, false, Bf, (short)0, C,
                                              false, false);
    // dist^2 = |p|^2 - 2 p.n + |n|^2 ; min over the 16 rows (M).
    // Column N = lane%16 is split across lanes l and l^16.
    float m = 3.0e38f;
#pragma unroll
    for (int r = 0; r < 8; ++r)
      m = fminf(m, p2[r] - 2.0f * C[r]);
    m = fminf(m, __shfl_xor(m, 16, 32));
    if (lane < 16) {
      float dd = sqrtf(fmaxf(m + n2, 0.0f));
      s_dist[n0 + lane] = dd;
      local_min = fminf(local_min, dd);
    }
  }

  // ---- block min of dist (softmax stabilizer) ----
  s_red[tid] = local_min;
  __syncthreads();
  for (int s = 128; s > 0; s >>= 1) {
    if (tid < s) s_red[tid] = fminf(s_red[tid], s_red[tid + s]);
    __syncthreads();
  }
  if (tid == 0) s_dmin = s_red[0];
  __syncthreads();
  const float dmin = s_dmin;

  // ---- phase 2: select the `cap` smallest by binary search on float bits ----
  // dist >= 0 so IEEE bit patterns are order-isomorphic to values.
  const int cap = *cap_p;
  const bool doSel = (cap > 0 && cap < N_);
  unsigned tbits = 0xFFFFFFFFu;   // no-selection case: everything < tbits
  int r_ties = N_;
  if (doSel) {
    unsigned lo = 0u, hi = 0x7F800000u;
    while (lo < hi) {
      const unsigned mid = lo + ((hi - lo) >> 1);
      if (tid == 0) s_cnt = 0u;
      __syncthreads();
      int c = 0;
#pragma unroll
      for (int i = 0; i < 16; ++i)
        c += (__float_as_uint(s_dist[tid * 16 + i]) <= mid) ? 1 : 0;
      atomicAdd(&s_cnt, (unsigned)c);
      __syncthreads();
      const unsigned cnt = s_cnt;
      __syncthreads();
      if (cnt >= (unsigned)cap) hi = mid; else lo = mid + 1u;
    }
    tbits = lo;
    // strict-less count -> how many ties at the threshold we may include
    if (tid == 0) s_cnt = 0u;
    __syncthreads();
    int c = 0;
#pragma unroll
    for (int i = 0; i < 16; ++i)
      c += (__float_as_uint(s_dist[tid * 16 + i]) < tbits) ? 1 : 0;
    atomicAdd(&s_cnt, (unsigned)c);
    __syncthreads();
    r_ties = cap - (int)s_cnt;
    __syncthreads();
  }

  // ---- tie ranking in index order (matches top_k tie-break): prefix scan ----
  int eq = 0;
#pragma unroll
  for (int i = 0; i < 16; ++i)
    eq += (__float_as_uint(s_dist[tid * 16 + i]) == tbits) ? 1 : 0;
  s_scan[tid] = eq;
  __syncthreads();
  for (int s = 1; s < 256; s <<= 1) {
    const int v = s_scan[tid];
    const int addv = (tid >= s) ? s_scan[tid - s] : 0;
    __syncthreads();
    s_scan[tid] = v + addv;
    __syncthreads();
  }
  const int eq_prefix = s_scan[tid] - eq;   // exclusive prefix

  // ---- GMM constants for this (b, w) ----
  if (tid == 0) {
    const float* pw = pi + (b * W_ + w) * K_;
    float mx = -3.0e38f;
    for (int k = 0; k < K_; ++k) mx = fmaxf(mx, pw[k]);
    float se = 0.0f;
    for (int k = 0; k < K_; ++k) se += expf(pw[k] - mx);
    s_lse = mx + logf(se);
  }
  __syncthreads();
  if (tid < K_) {
    const int base = ((b * W_ + w) * K_ + tid) * 2;
    const float m0 = mu[base],    m1 = mu[base + 1];
    const float g0 = sigma[base], g1 = sigma[base + 1];
    s_mu0[tid] = m0; s_mu1[tid] = m1;
    s_is0[tid] = 1.0f / g0; s_is1[tid] = 1.0f / g1;
    s_ck[tid] = (pi[(b * W_ + w) * K_ + tid] - s_lse)
              - logf(g0) - logf(g1) - LOG2PI_;
  }
  __syncthreads();

  // ---- phase 3: weighted GMM log-prob over the selected set ----
  const int flip = *flip_p;
  float S = 0.0f, Z = 0.0f;
  int eqc = 0;
  for (int i = 0; i < 16; ++i) {
    const int n = tid * 16 + i;
    const float dd = s_dist[n];
    const unsigned bb = __float_as_uint(dd);
    bool sel = false;
    if (bb < tbits) sel = true;
    else if (bb == tbits) { sel = (eq_prefix + eqc) < r_ties; ++eqc; }
    if (sel) {
      const float2 q = negv[b * N_ + n];
      const float x0 = flip ? q.y : q.x;   // wh-flip matters vs. mu/sigma
      const float x1 = flip ? q.x : q.y;
      float a[K_];
      float amax = -3.0e38f;
#pragma unroll
      for (int k = 0; k < K_; ++k) {
        const float z0 = (x0 - s_mu0[k]) * s_is0[k];
        const float z1 = (x1 - s_mu1[k]) * s_is1[k];
        a[k] = s_ck[k] - 0.5f * (z0 * z0 + z1 * z1);
        amax = fmaxf(amax, a[k]);
      }
      float se = 0.0f;
#pragma unroll
      for (int k = 0; k < K_; ++k) se += expf(a[k] - amax);
      const float lp = amax + logf(se);
      const float e = expf(dmin - dd);     // un-normalized softmax weight
      Z += e;
      S += e * lp;
    }
  }

  // ---- block sums of S and Z; partial[bw] = S/Z ----
  s_red[tid] = S;
  __syncthreads();
  for (int s = 128; s > 0; s >>= 1) {
    if (tid < s) s_red[tid] += s_red[tid + s];
    __syncthreads();
  }
  if (tid == 0) s_S = s_red[0];
  __syncthreads();
  s_red[tid] = Z;
  __syncthreads();
  for (int s = 128; s > 0; s >>= 1) {
    if (tid < s) s_red[tid] += s_red[tid + s];
    __syncthreads();
  }
  if (tid == 0) partial[bw] = s_S / s_red[0];
}

// Deterministic finalize: out[b] = -(1/W) * sum_w partial[b,w]
__global__ void mdn_loss_final(const float* __restrict__ partial,
                               float* __restrict__ out)
{
  const int b = blockIdx.x * blockDim.x + threadIdx.x;
  if (b < BS_) {
    float s = 0.0f;
    for (int w = 0; w < W_; ++w) s += partial[b * W_ + w];
    out[b] = -s * (1.0f / (float)W_);
  }
}

extern "C" void kernel_launch(void* const* d_in, const int* in_sizes, int n_in,
                              void* d_out, int out_size, void* d_ws, size_t ws_size,
                              hipStream_t stream) {
  (void)in_sizes; (void)n_in; (void)out_size; (void)ws_size;
  const float* mu    = (const float*)d_in[0];
  const float* sigma = (const float*)d_in[1];
  const float* pi    = (const float*)d_in[2];
  const float* neg   = (const float*)d_in[3];
  const float* pos   = (const float*)d_in[4];
  const int*   flip  = (const int*)d_in[5];
  const int*   cap   = (const int*)d_in[6];
  float* partial = (float*)d_ws;          // 512 floats of scratch
  float* out = (float*)d_out;             // 64 floats

  mdn_loss_main<<<BS_ * W_, 256, 0, stream>>>(mu, sigma, pi, neg, pos,
                                              flip, cap, partial);
  mdn_loss_final<<<1, BS_, 0, stream>>>(partial, out);
}